// OneLayerMLP_51496657879062
// MI455X (gfx1250) — compile-verified
//
#include <hip/hip_runtime.h>

typedef __attribute__((ext_vector_type(2))) float v2f;
typedef __attribute__((ext_vector_type(8))) float v8f;

#define VOCAB 50000
#define EMB_D 512
#define HID   2048
#define NCLS  151
#define NCLSP 192   // NCLS padded up to a multiple of 64 (one wave strip)
#define BATCH 2048
#define SEQ   128

// ---------------------------------------------------------------------------
// Stage 1: masked-average pooling of embedding rows.
// One block per batch row; 128 threads, each owning 4 consecutive floats of D.
// summed == plain sum (masked-out rows are exactly the all-zero rows), so we
// accumulate unconditionally and only track per-token "row is nonzero" bits.
// ---------------------------------------------------------------------------
__global__ __launch_bounds__(128) void pool_avg_kernel(
    const int* __restrict__ ids, const float* __restrict__ emb,
    float* __restrict__ avg) {
  __shared__ int s_ids[SEQ];
  __shared__ unsigned s_nz[4];
  const int b = blockIdx.x;
  const int t = threadIdx.x;
  if (t < SEQ) s_ids[t] = ids[(size_t)b * SEQ + t];
  if (t < 4)   s_nz[t] = 0u;
  __syncthreads();

  float4 acc = make_float4(0.f, 0.f, 0.f, 0.f);
  unsigned nzm[4] = {0u, 0u, 0u, 0u};
  for (int s = 0; s < SEQ; ++s) {
    const int id = s_ids[s];
    const float4 v = ((const float4*)(emb + (size_t)id * EMB_D))[t];  // b128
    acc.x += v.x; acc.y += v.y; acc.z += v.z; acc.w += v.w;
    const bool nz = (v.x != 0.f) | (v.y != 0.f) | (v.z != 0.f) | (v.w != 0.f);
    nzm[s >> 5] |= ((unsigned)nz) << (s & 31);
  }
#pragma unroll
  for (int w = 0; w < 4; ++w) atomicOr(&s_nz[w], nzm[w]);
  __syncthreads();
  const int cnt = __popc(s_nz[0]) + __popc(s_nz[1]) +
                  __popc(s_nz[2]) + __popc(s_nz[3]);
  const float scale = (cnt > 0) ? (1.0f / (float)cnt) : 0.0f;
  acc.x *= scale; acc.y *= scale; acc.z *= scale; acc.w *= scale;
  ((float4*)(avg + (size_t)b * EMB_D))[t] = acc;
}

// ---------------------------------------------------------------------------
// Weight pre-pack: out[n][k] = (n < Nout) ? in[k][n] : 0   (n-major, k contig)
// Makes every B-fragment load in the GEMM a contiguous b64 and removes all
// bounds checks from the GEMM inner loop. Total traffic ~5.5 MB, negligible.
// ---------------------------------------------------------------------------
__global__ __launch_bounds__(256) void transpose_pad_kernel(
    const float* __restrict__ in, float* __restrict__ out,
    int K, int Nout, int Npad) {
  const int idx = blockIdx.x * 256 + threadIdx.x;
  if (idx >= K * Npad) return;
  const int n = idx / K;
  const int k = idx - n * K;
  out[(size_t)n * K + k] = (n < Nout) ? in[(size_t)k * Nout + n] : 0.0f;
}

// ---------------------------------------------------------------------------
// f32 WMMA GEMM, A row-major [M,K], Bt n-major [Npad,K] (= B transposed).
// Each wave owns a 16(M) x 64(N) strip: 4 accumulator tiles, A fragment
// reused across the 4 N-tiles. K-step = 4 (V_WMMA_F32_16X16X4_F32).
// Fragment layout (ISA 7.12.2, 32-bit): v2f element j holds K = 2*(lane/16)+j;
// A row = lane%16, B col = lane%16 -> both fragments are contiguous b64 loads.
// C/D tile: VGPR r -> M = r + 8*(lane/16), N = lane%16.
// Inner loop is branch-free; EXEC is all-1s throughout.
// ---------------------------------------------------------------------------
__global__ __launch_bounds__(256) void wmma_gemm_tn(
    const float* __restrict__ A, const float* __restrict__ Bt,
    const float* __restrict__ bias, float* __restrict__ C,
    int K, int Nout, int ldc, int relu, int nn, int nstrips) {
  const int wave = threadIdx.x >> 5;
  const int lane = threadIdx.x & 31;
  const int strip = blockIdx.x * 8 + wave;
  if (strip >= nstrips) return;  // wave-uniform
  const int m0 = (strip / nn) << 4;
  const int n0 = (strip % nn) << 6;
  const int hi = lane >> 4;
  const int lm = lane & 15;

  v8f acc0 = {}, acc1 = {}, acc2 = {}, acc3 = {};
  const float* ap  = A  + (size_t)(m0 + lm) * K + 2 * hi;
  const float* bp0 = Bt + (size_t)(n0 + lm) * K + 2 * hi;
  const float* bp1 = bp0 + (size_t)16 * K;
  const float* bp2 = bp0 + (size_t)32 * K;
  const float* bp3 = bp0 + (size_t)48 * K;

#pragma unroll 2
  for (int k = 0; k < K; k += 4) {
    const v2f a  = *(const v2f*)(ap  + k);   // b64
    const v2f b0 = *(const v2f*)(bp0 + k);   // b64
    const v2f b1 = *(const v2f*)(bp1 + k);
    const v2f b2 = *(const v2f*)(bp2 + k);
    const v2f b3 = *(const v2f*)(bp3 + k);
    acc0 = __builtin_amdgcn_wmma_f32_16x16x4_f32(false, a, false, b0, (short)0, acc0, false, false);
    acc1 = __builtin_amdgcn_wmma_f32_16x16x4_f32(false, a, false, b1, (short)0, acc1, false, false);
    acc2 = __builtin_amdgcn_wmma_f32_16x16x4_f32(false, a, false, b2, (short)0, acc2, false, false);
    acc3 = __builtin_amdgcn_wmma_f32_16x16x4_f32(false, a, false, b3, (short)0, acc3, false, false);
  }

  // Epilogue: bias + optional ReLU; output bound (N=151) guarded only here.
  auto store_tile = [&](const v8f& tacc, int tn) {
    const int n = n0 + tn * 16 + lm;
    if (n < Nout) {
      const float bv = bias[n];
#pragma unroll
      for (int r = 0; r < 8; ++r) {
        float v = tacc[r] + bv;
        if (relu) v = fmaxf(v, 0.0f);
        C[(size_t)(m0 + r + 8 * hi) * ldc + n] = v;
      }
    }
  };
  store_tile(acc0, 0);
  store_tile(acc1, 1);
  store_tile(acc2, 2);
  store_tile(acc3, 3);
}

// ---------------------------------------------------------------------------
// Launch: pack W1t/W2t -> pool -> GEMM1(relu) -> GEMM2
// Workspace layout (floats):
//   avg  [2048*512]   =  4 MB
//   h    [2048*2048]  = 16 MB
//   W1t  [2048*512]   =  4 MB
//   W2t  [192*2048]   =  1.5 MB        (total ~25.5 MB)
// ---------------------------------------------------------------------------
extern "C" void kernel_launch(void* const* d_in, const int* in_sizes, int n_in,
                              void* d_out, int out_size, void* d_ws, size_t ws_size,
                              hipStream_t stream) {
  const int*   ids = (const int*)d_in[0];
  const float* emb = (const float*)d_in[1];
  const float* W1  = (const float*)d_in[2];
  const float* b1  = (const float*)d_in[3];
  const float* W2  = (const float*)d_in[4];
  const float* b2  = (const float*)d_in[5];
  float* out = (float*)d_out;

  float* avg = (float*)d_ws;
  float* h   = avg + (size_t)BATCH * EMB_D;
  float* w1t = h   + (size_t)BATCH * HID;
  float* w2t = w1t + (size_t)HID * EMB_D;

  // Weight pre-pack (independent of pool; all serialized on stream).
  {
    const int tot1 = EMB_D * HID;                 // W1t: [HID][EMB_D]
    transpose_pad_kernel<<<(tot1 + 255) / 256, 256, 0, stream>>>(
        W1, w1t, EMB_D, HID, HID);
    const int tot2 = HID * NCLSP;                 // W2t: [192][HID]
    transpose_pad_kernel<<<(tot2 + 255) / 256, 256, 0, stream>>>(
        W2, w2t, HID, NCLS, NCLSP);
  }

  pool_avg_kernel<<<BATCH, 128, 0, stream>>>(ids, emb, avg);

  // GEMM1: [2048,512] @ [512,2048] + b1, ReLU
  {
    const int nn = HID / 64;                      // 32
    const int strips = (BATCH / 16) * nn;         // 4096
    wmma_gemm_tn<<<(strips + 7) / 8, 256, 0, stream>>>(
        avg, w1t, b1, h, EMB_D, HID, HID, /*relu=*/1, nn, strips);
  }

  // GEMM2: [2048,2048] @ [2048,151] + b2
  {
    const int nn = NCLSP / 64;                    // 3
    const int strips = (BATCH / 16) * nn;         // 384
    wmma_gemm_tn<<<(strips + 7) / 8, 256, 0, stream>>>(
        h, w2t, b2, out, HID, NCLS, NCLS, /*relu=*/0, nn, strips);
  }
}